// RNNModel_73400991088715
// MI455X (gfx1250) — compile-verified
//
#include <hip/hip_runtime.h>
#include <math.h>

namespace {
constexpr int BB = 64;    // batch
constexpr int TT = 1024;  // timesteps
constexpr int II = 256;   // input size (layer 0)
constexpr int HH = 512;   // hidden size
constexpr int BH = BB * HH;            // 32768
constexpr int KCH = 32;                // K chunk per LDS stage
constexpr int MROWS = 32;              // batch rows per block (gridDim.y = 2)
constexpr int LDA = KCH + 4;           // padded LDS A row stride (floats)
constexpr int LDG = 132;               // padded LDS gate row stride (floats)
constexpr int BUF = MROWS * LDA + (KCH / 4) * 128 * 4;  // 5248 floats per stage buffer
constexpr int LDS_FLOATS = 2 * BUF;    // 10496 floats (~42KB); >= MROWS*LDG (4224)
}

typedef __attribute__((ext_vector_type(2))) float v2f;
typedef __attribute__((ext_vector_type(8))) float v8f;
typedef int v4i __attribute__((vector_size(16)));   // matches builtin param type

#define AS1 __attribute__((address_space(1)))
#define AS3 __attribute__((address_space(3)))

#if defined(__has_builtin)
#if __has_builtin(__builtin_amdgcn_global_load_async_to_lds_b128)
#define HAVE_ASYNC_LDS 1
#endif
#endif
#ifndef HAVE_ASYNC_LDS
#define HAVE_ASYNC_LDS 0
#endif

// 16B global -> LDS copy; async DMA path on gfx1250 (ASYNCcnt), sync fallback.
__device__ __forceinline__ void copy16(const float* g, float* l) {
#if HAVE_ASYNC_LDS
  __builtin_amdgcn_global_load_async_to_lds_b128(
      (AS1 v4i*)(AS1 void*)g, (AS3 v4i*)(AS3 void*)l, 0, 0);
#else
  *(float4*)l = *(const float4*)g;
#endif
}

__device__ __forceinline__ void waitAsync0() {
#if HAVE_ASYNC_LDS
#if __has_builtin(__builtin_amdgcn_s_wait_asynccnt)
  __builtin_amdgcn_s_wait_asynccnt(0);
#else
  asm volatile("s_wait_asynccnt 0" ::: "memory");
#endif
#endif
}

// Stage one K-chunk (A: 32x32 activations, B: 128x32 of W^T) into an LDS buffer.
__device__ __forceinline__ void issueChunk(
    float* buf, int tid, int mbase, int j0, int kc, int Kin,
    const float* __restrict__ x, int xStride,
    const float* __restrict__ hprev,
    const float* __restrict__ Wih, const float* __restrict__ Whh)
{
  const bool fromX    = (kc < Kin);
  const float* aSrc   = fromX ? x : hprev;
  const int   aStride = fromX ? xStride : HH;
  const float* wSrc   = fromX ? Wih : Whh;
  const int   wStride = fromX ? Kin : HH;
  const int   kloc    = fromX ? kc : (kc - Kin);

  // A tile: 32 rows x 32 k = 256 float4, 1 per thread, coalesced.
  {
    int m = tid >> 3, f4 = tid & 7;
    copy16(aSrc + (size_t)(mbase + m) * aStride + kloc + f4 * 4,
           buf + m * LDA + f4 * 4);
  }
  // B tile (W^T): 128 gate-cols x 32 k = 1024 float4, 4 per thread,
  // stored k-interleaved [k/4][128][4] so fragments are contiguous b64 reads.
  float* bufB = buf + MROWS * LDA;
#pragma unroll
  for (int r = 0; r < 4; ++r) {
    int q = tid + r * 256;
    int nn = q >> 3, f4 = q & 7;
    int wrow = (nn >> 5) * HH + j0 + (nn & 31);
    copy16(wSrc + (size_t)wrow * wStride + kloc + f4 * 4,
           bufB + (f4 * 128 + nn) * 4);
  }
}

// One LSTM timestep for one layer.
// gates[b][n] = sum_k [x_t | h_prev][b][k] * [W_ih;W_hh]^T[k][n] + bias[n]
// Block: 32-wide hidden slice across all 4 gates (128 cols) x 32 batch rows.
// Software-pipelined: async-load chunk i+1 to LDS while WMMAs run on chunk i.
__global__ __launch_bounds__(256) void lstm_step_kernel(
    const float* __restrict__ x, int xStride, int Kin,
    const float* __restrict__ Wih, const float* __restrict__ Whh,
    const float* __restrict__ bias,      // combined b_ih + b_hh, [4H]
    const float* __restrict__ hprev, const float* __restrict__ cprev,
    float* __restrict__ hnext, float* __restrict__ cnext,
    float* __restrict__ yout, int yStride)
{
  __shared__ float lds[LDS_FLOATS];

  const int tid      = threadIdx.x;
  const int wave     = tid >> 5;
  const int lane     = tid & 31;
  const int lane16   = lane & 15;
  const int laneHalf = lane >> 4;            // 0: K=0,1 / M+0 ; 1: K=2,3 / M+8
  const int j0       = blockIdx.x * 32;      // hidden-index slice
  const int mbase    = blockIdx.y * MROWS;   // batch-row half
  const int gate     = wave >> 1;            // 0..3 = i,f,g,o
  const int nLds     = gate * 32 + (wave & 1) * 16;  // col base within 128

  v8f acc[2] = {};

  const int nCh = (Kin + HH) / KCH;          // 24 or 32 chunks
  issueChunk(lds, tid, mbase, j0, 0, Kin, x, xStride, hprev, Wih, Whh);

  int cur = 0;
  for (int ic = 0; ic < nCh; ++ic) {
    waitAsync0();          // my chunk-ic DMA done
    __syncthreads();       // all waves' DMA done + previous buffer free
    if (ic + 1 < nCh)
      issueChunk(lds + (cur ^ 1) * BUF, tid, mbase, j0, (ic + 1) * KCH, Kin,
                 x, xStride, hprev, Wih, Whh);

    const float* ldsA = lds + cur * BUF;
    const float* ldsB = ldsA + MROWS * LDA;
    // 8 k-steps of V_WMMA_F32_16X16X4_F32, 2 M-tiles per wave.
#pragma unroll
    for (int ks = 0; ks < 8; ++ks) {
      v2f bf = *(const v2f*)(ldsB + (ks * 128 + nLds + lane16) * 4 + 2 * laneHalf);
#pragma unroll
      for (int mt = 0; mt < 2; ++mt) {
        v2f af = *(const v2f*)(ldsA + (mt * 16 + lane16) * LDA + ks * 4 + 2 * laneHalf);
        acc[mt] = __builtin_amdgcn_wmma_f32_16x16x4_f32(
            false, af, false, bf, (short)0, acc[mt], false, false);
      }
    }
    cur ^= 1;
  }
  __syncthreads();

  // Scatter pre-activation gates to LDS [MROWS][LDG] (C layout: vgpr r -> M=r / r+8).
#pragma unroll
  for (int mt = 0; mt < 2; ++mt) {
#pragma unroll
    for (int r = 0; r < 8; ++r) {
      int m = mt * 16 + laneHalf * 8 + r;
      lds[m * LDG + nLds + lane16] = acc[mt][r];
    }
  }
  __syncthreads();

  // Cell update: 32 rows x 32 cols = 1024 elements, 4 per thread.
#pragma unroll
  for (int r = 0; r < 4; ++r) {
    int p     = tid + r * 256;
    int bl    = p >> 5;
    int jj    = p & 31;
    int bglob = mbase + bl;
    int jg    = j0 + jj;
    float iv = lds[bl * LDG +  0 + jj] + bias[0 * HH + jg];
    float fv = lds[bl * LDG + 32 + jj] + bias[1 * HH + jg];
    float gv = lds[bl * LDG + 64 + jj] + bias[2 * HH + jg];
    float ov = lds[bl * LDG + 96 + jj] + bias[3 * HH + jg];
    float ig = 1.0f / (1.0f + __expf(-iv));
    float fg = 1.0f / (1.0f + __expf(-fv));
    float gg = tanhf(gv);
    float og = 1.0f / (1.0f + __expf(-ov));
    size_t sidx = (size_t)bglob * HH + jg;
    float c = fg * cprev[sidx] + ig * gg;
    float h = og * tanhf(c);
    cnext[sidx] = c;
    hnext[sidx] = h;
    yout[(size_t)bglob * yStride + jg] = h;
  }
}

// Copy h0/c0 into ping-pong state buffers and precombine biases.
__global__ __launch_bounds__(256) void lstm_init_kernel(
    const float* __restrict__ h0, const float* __restrict__ c0,
    const float* __restrict__ bih0, const float* __restrict__ bhh0,
    const float* __restrict__ bih1, const float* __restrict__ bhh1,
    float* __restrict__ st, float* __restrict__ bias)
{
  int i = blockIdx.x * 256 + threadIdx.x;
  if (i < 2 * BH) {
    int l = i / BH, r = i - l * BH;
    st[l * 4 * BH + 0 * BH + r] = h0[i];   // h buf 0
    st[l * 4 * BH + 2 * BH + r] = c0[i];   // c buf 0
  }
  if (i < 4 * HH) {
    bias[i]          = bih0[i] + bhh0[i];
    bias[4 * HH + i] = bih1[i] + bhh1[i];
  }
}

// T is even, so final state lands in ping buffer 0.
__global__ __launch_bounds__(256) void lstm_final_kernel(
    const float* __restrict__ st, float* __restrict__ hn, float* __restrict__ cn)
{
  int i = blockIdx.x * 256 + threadIdx.x;
  if (i < 2 * BH) {
    int l = i / BH, r = i - l * BH;
    hn[i] = st[l * 4 * BH + 0 * BH + r];
    cn[i] = st[l * 4 * BH + 2 * BH + r];
  }
}

extern "C" void kernel_launch(void* const* d_in, const int* in_sizes, int n_in,
                              void* d_out, int out_size, void* d_ws, size_t ws_size,
                              hipStream_t stream) {
  const float* inputs = (const float*)d_in[0];
  const float* h0     = (const float*)d_in[1];
  const float* c0     = (const float*)d_in[2];
  const float* Wih0   = (const float*)d_in[3];
  const float* Whh0   = (const float*)d_in[4];
  const float* bih0   = (const float*)d_in[5];
  const float* bhh0   = (const float*)d_in[6];
  const float* Wih1   = (const float*)d_in[7];
  const float* Whh1   = (const float*)d_in[8];
  const float* bih1   = (const float*)d_in[9];
  const float* bhh1   = (const float*)d_in[10];
  float* out = (float*)d_out;
  float* ws  = (float*)d_ws;

  // Workspace layout (floats):
  float* y0   = ws;                            // [B,T,H] layer-0 output
  float* st   = y0 + (size_t)BB * TT * HH;     // 2 layers * {h0,h1,c0,c1} * BH
  float* bias = st + (size_t)2 * 4 * BH;       // 2 * 4H combined biases

  lstm_init_kernel<<<(2 * BH + 255) / 256, 256, 0, stream>>>(
      h0, c0, bih0, bhh0, bih1, bhh1, st, bias);

  dim3 grid(16, 2);  // 16 hidden slices x 2 batch halves

  // Layer 0: x_t = inputs[:, t, :]  (row stride T*IN)
  for (int t = 0; t < TT; ++t) {
    int pi = t & 1, po = pi ^ 1;
    lstm_step_kernel<<<grid, 256, 0, stream>>>(
        inputs + (size_t)t * II, TT * II, II,
        Wih0, Whh0, bias,
        st + pi * BH, st + 2 * BH + pi * BH,
        st + po * BH, st + 2 * BH + po * BH,
        y0 + (size_t)t * HH, TT * HH);
  }
  // Layer 1: x_t = y0[:, t, :]  (row stride T*H), write y1 into d_out
  for (int t = 0; t < TT; ++t) {
    int pi = t & 1, po = pi ^ 1;
    float* stl = st + 4 * BH;
    lstm_step_kernel<<<grid, 256, 0, stream>>>(
        y0 + (size_t)t * HH, TT * HH, HH,
        Wih1, Whh1, bias + 4 * HH,
        stl + pi * BH, stl + 2 * BH + pi * BH,
        stl + po * BH, stl + 2 * BH + po * BH,
        out + (size_t)t * HH, TT * HH);
  }
  lstm_final_kernel<<<(2 * BH + 255) / 256, 256, 0, stream>>>(
      st, out + (size_t)BB * TT * HH, out + (size_t)BB * TT * HH + 2 * BH);

  (void)in_sizes; (void)n_in; (void)out_size; (void)ws_size;
}